// DenseGAT_21045339750899
// MI455X (gfx1250) — compile-verified
//
#include <hip/hip_runtime.h>
#include <math.h>

// ---------------------------------------------------------------------------
// 2-layer GAT (PyG GATConv semantics) for MI455X / gfx1250.
// Dense projections via V_WMMA_F32_16X16X4_F32 (exact fp32 matrix path).
// Edge softmax/aggregation via L2-resident fp32 atomics (all hot arrays < 96MB,
// well inside the 192MB L2).
// ---------------------------------------------------------------------------

typedef __attribute__((ext_vector_type(2))) float v2f;
typedef __attribute__((ext_vector_type(8))) float v8f;

#define N_NODES 50000
#define N_EDGES 800000
#define IN_CH   128
#define HID     16
#define HEADS   8
#define OUT_CH  16
#define NEG_SLOPE 0.2f

// ---- ordered-uint mapping for float atomic max ----------------------------
static __device__ __forceinline__ unsigned f2ord(float f) {
    unsigned u = __float_as_uint(f);
    return (u & 0x80000000u) ? ~u : (u | 0x80000000u);
}
static __device__ __forceinline__ float ord2f(unsigned u) {
    unsigned v = (u & 0x80000000u) ? (u & 0x7fffffffu) : ~u;
    return __uint_as_float(v);
}
#define ORD_NEG_INF 0x007fffffu   // f2ord(-inf)

static __device__ __forceinline__ float elu1(float x) {
    return x > 0.0f ? x : expm1f(x);
}
static __device__ __forceinline__ float lrelu(float x) {
    return x >= 0.0f ? x : NEG_SLOPE * x;
}

// ---------------------------------------------------------------------------
// Workspace init: zero accumulators, seed segment-max with ord(-inf).
// ---------------------------------------------------------------------------
__global__ void k_init(float* __restrict__ out1, float* __restrict__ out2,
                       unsigned* __restrict__ m1raw, float* __restrict__ den1,
                       unsigned* __restrict__ m2raw, float* __restrict__ den2) {
    int idx = blockIdx.x * blockDim.x + threadIdx.x;
    if (idx < N_NODES * IN_CH) out1[idx] = 0.0f;
    if (idx < N_NODES * OUT_CH) out2[idx] = 0.0f;
    if (idx < N_NODES * HEADS) { m1raw[idx] = ORD_NEG_INF; den1[idx] = 0.0f; }
    if (idx < N_NODES) { m2raw[idx] = ORD_NEG_INF; den2[idx] = 0.0f; }
}

// ---------------------------------------------------------------------------
// Edge-attr projection is rank-1: al_e[e,h] = ew[e] * sum_c We[h*C+c]*att_e[h,c]
// Fold into per-head constants. cws[0..7] = layer1 heads, cws[8] = layer2.
// ---------------------------------------------------------------------------
__global__ void k_edge_consts(const float* __restrict__ We1, const float* __restrict__ ae1,
                              const float* __restrict__ We2, const float* __restrict__ ae2,
                              float* __restrict__ cws) {
    int t = threadIdx.x;
    if (t < HEADS) {
        float s = 0.0f;
        for (int c = 0; c < HID; ++c) s += We1[t * HID + c] * ae1[t * HID + c];
        cws[t] = s;
    } else if (t == HEADS) {
        float s = 0.0f;
        for (int c = 0; c < OUT_CH; ++c) s += We2[c] * ae2[c];
        cws[HEADS] = s;
    }
}

// ---------------------------------------------------------------------------
// fp32 WMMA GEMM: C[M x N] = A[M x K] @ B[K x N], one wave per 16x16 tile,
// K stepped by 4 via V_WMMA_F32_16X16X4_F32. 4 waves per 128-thread block.
// A 16x4 layout: lanes 0-15 hold M=0..15; VGPR0=K{0|2}, VGPR1=K{1|3} per half.
// ---------------------------------------------------------------------------
__global__ void k_gemm_f32(const float* __restrict__ A, const float* __restrict__ B,
                           float* __restrict__ C, int M, int K, int N) {
    int tilesN = N >> 4;
    int tile = blockIdx.x * (blockDim.x >> 5) + (threadIdx.x >> 5);
    int totalTiles = (M >> 4) * tilesN;
    if (tile >= totalTiles) return;
    int tm = tile / tilesN, tn = tile % tilesN;
    int lane = threadIdx.x & 31;
    int l15  = lane & 15;
    int kh   = (lane >> 4) << 1;          // 0 (lanes 0-15) or 2 (lanes 16-31)
    const float* arow = A + (size_t)(tm * 16 + l15) * K;
    const float* bcol = B + tn * 16 + l15;
    v8f acc = {};
    for (int k = 0; k < K; k += 4) {
        v2f a, b;
        a.x = arow[k + kh];
        a.y = arow[k + kh + 1];
        b.x = bcol[(size_t)(k + kh) * N];
        b.y = bcol[(size_t)(k + kh + 1) * N];
        acc = __builtin_amdgcn_wmma_f32_16x16x4_f32(false, a, false, b,
                                                    (short)0, acc, false, false);
    }
    int rbase = tm * 16 + ((lane >> 4) << 3);   // rows r / r+8 across lane halves
    int col = tn * 16 + l15;
    for (int r = 0; r < 8; ++r)
        C[(size_t)(rbase + r) * N + col] = acc[r];
}

// Same GEMM but A element = elu(Ain + bias[k])  (layer-2 input activation).
__global__ void k_gemm_elu_f32(const float* __restrict__ Ain, const float* __restrict__ bias,
                               const float* __restrict__ B, float* __restrict__ C,
                               int M, int K, int N) {
    int tilesN = N >> 4;
    int tile = blockIdx.x * (blockDim.x >> 5) + (threadIdx.x >> 5);
    int totalTiles = (M >> 4) * tilesN;
    if (tile >= totalTiles) return;
    int tm = tile / tilesN, tn = tile % tilesN;
    int lane = threadIdx.x & 31;
    int l15  = lane & 15;
    int kh   = (lane >> 4) << 1;
    const float* arow = Ain + (size_t)(tm * 16 + l15) * K;
    const float* bcol = B + tn * 16 + l15;
    v8f acc = {};
    for (int k = 0; k < K; k += 4) {
        v2f a, b;
        a.x = elu1(arow[k + kh]     + bias[k + kh]);
        a.y = elu1(arow[k + kh + 1] + bias[k + kh + 1]);
        b.x = bcol[(size_t)(k + kh) * N];
        b.y = bcol[(size_t)(k + kh + 1) * N];
        acc = __builtin_amdgcn_wmma_f32_16x16x4_f32(false, a, false, b,
                                                    (short)0, acc, false, false);
    }
    int rbase = tm * 16 + ((lane >> 4) << 3);
    int col = tn * 16 + l15;
    for (int r = 0; r < 8; ++r)
        C[(size_t)(rbase + r) * N + col] = acc[r];
}

// ---------------------------------------------------------------------------
// Per-node attention scalars: al_s[n,h] = <h[n,h,:], att_src[h,:]>  (etc.)
// ---------------------------------------------------------------------------
__global__ void k_node_attn(const float* __restrict__ h, const float* __restrict__ a_s,
                            const float* __restrict__ a_d, float* __restrict__ al_s,
                            float* __restrict__ al_d, int heads, int ch) {
    int idx = blockIdx.x * blockDim.x + threadIdx.x;
    if (idx >= N_NODES * heads) return;
    int n = idx / heads, hd = idx % heads;
    const float* hp = h + (size_t)n * heads * ch + hd * ch;
    float s = 0.0f, d = 0.0f;
    for (int c = 0; c < ch; ++c) {
        float v = hp[c];
        s += v * a_s[hd * ch + c];
        d += v * a_d[hd * ch + c];
    }
    al_s[idx] = s;
    al_d[idx] = d;
}

// ---------------------------------------------------------------------------
// Edge pass 1: raw score alpha = leaky_relu(al_s[src]+al_d[dst]+ew*c_e),
// store alpha, atomic segment-max over dst (ordered-uint trick).
// ---------------------------------------------------------------------------
__global__ void k_edge_score(const int* __restrict__ ei, const float* __restrict__ ew,
                             const float* __restrict__ al_s, const float* __restrict__ al_d,
                             const float* __restrict__ cws, int cofs,
                             float* __restrict__ alpha, unsigned* __restrict__ mraw,
                             int heads) {
    int idx = blockIdx.x * blockDim.x + threadIdx.x;
    if (idx >= N_EDGES * heads) return;
    int e = idx / heads, hd = idx % heads;
    int src = ei[e], dst = ei[N_EDGES + e];
    float a = lrelu(al_s[src * heads + hd] + al_d[dst * heads + hd] + ew[e] * cws[cofs + hd]);
    alpha[idx] = a;
    atomicMax(&mraw[dst * heads + hd], f2ord(a));
}

// Max fixup: decode ordered-uint; non-finite (empty segment) -> 0, per reference.
__global__ void k_max_fix(unsigned* __restrict__ mraw, float* __restrict__ m, int count) {
    int idx = blockIdx.x * blockDim.x + threadIdx.x;
    if (idx >= count) return;
    float v = ord2f(mraw[idx]);
    m[idx] = isfinite(v) ? v : 0.0f;
}

// Edge pass 2: ex = exp(alpha - m[dst]); overwrite alpha with ex; atomic denom.
__global__ void k_edge_exp(const int* __restrict__ ei, float* __restrict__ alpha,
                           const float* __restrict__ m, float* __restrict__ den, int heads) {
    int idx = blockIdx.x * blockDim.x + threadIdx.x;
    if (idx >= N_EDGES * heads) return;
    int e = idx / heads, hd = idx % heads;
    int dst = ei[N_EDGES + e];
    float ex = expf(alpha[idx] - m[dst * heads + hd]);
    alpha[idx] = ex;
    atomicAdd(&den[dst * heads + hd], ex);
}

// Edge pass 3: out[dst, k] += h[src, k] * ex/(den[dst]+1e-16). One thread per
// (edge, channel); gathers hit L2 (h fits on-chip).
__global__ void k_edge_aggr(const int* __restrict__ ei, const float* __restrict__ h,
                            const float* __restrict__ alpha, const float* __restrict__ den,
                            float* __restrict__ out, int heads, int ch) {
    int F = heads * ch;
    long long idx = (long long)blockIdx.x * blockDim.x + threadIdx.x;
    if (idx >= (long long)N_EDGES * F) return;
    int e = (int)(idx / F), k = (int)(idx % F);
    int hd = k / ch;
    int src = ei[e], dst = ei[N_EDGES + e];
    float w = alpha[(size_t)e * heads + hd] / (den[(size_t)dst * heads + hd] + 1e-16f);
    atomicAdd(&out[(size_t)dst * F + k], h[(size_t)src * F + k] * w);
}

// Final: log_softmax over 16 channels (+ zero bias kept for fidelity).
__global__ void k_logsoftmax(const float* __restrict__ out2, const float* __restrict__ b2,
                             float* __restrict__ y) {
    int n = blockIdx.x * blockDim.x + threadIdx.x;
    if (n >= N_NODES) return;
    float v[OUT_CH];
    float mx = -INFINITY;
    for (int c = 0; c < OUT_CH; ++c) {
        v[c] = out2[(size_t)n * OUT_CH + c] + b2[c];
        mx = fmaxf(mx, v[c]);
    }
    float s = 0.0f;
    for (int c = 0; c < OUT_CH; ++c) s += expf(v[c] - mx);
    float lse = mx + logf(s);
    for (int c = 0; c < OUT_CH; ++c) y[(size_t)n * OUT_CH + c] = v[c] - lse;
}

// ---------------------------------------------------------------------------
extern "C" void kernel_launch(void* const* d_in, const int* in_sizes, int n_in,
                              void* d_out, int out_size, void* d_ws, size_t ws_size,
                              hipStream_t stream) {
    (void)in_sizes; (void)n_in; (void)out_size; (void)ws_size;
    const float* x    = (const float*)d_in[0];
    const int*   ei   = (const int*)  d_in[1];
    const float* ew   = (const float*)d_in[2];
    const float* W1   = (const float*)d_in[3];
    const float* as1  = (const float*)d_in[4];
    const float* ad1  = (const float*)d_in[5];
    const float* We1  = (const float*)d_in[6];
    const float* ae1  = (const float*)d_in[7];
    const float* b1   = (const float*)d_in[8];
    const float* W2   = (const float*)d_in[9];
    const float* as2  = (const float*)d_in[10];
    const float* ad2  = (const float*)d_in[11];
    const float* We2  = (const float*)d_in[12];
    const float* ae2  = (const float*)d_in[13];
    const float* b2   = (const float*)d_in[14];
    float* y = (float*)d_out;

    // workspace layout (floats)
    float* w = (float*)d_ws;
    size_t o = 0;
    float*    h1     = w + o;               o += (size_t)N_NODES * IN_CH;   // 6.4M
    float*    als1   = w + o;               o += (size_t)N_NODES * HEADS;
    float*    ald1   = w + o;               o += (size_t)N_NODES * HEADS;
    unsigned* m1raw  = (unsigned*)(w + o);
    float*    m1     = w + o;               o += (size_t)N_NODES * HEADS;
    float*    den1   = w + o;               o += (size_t)N_NODES * HEADS;
    float*    alpha1 = w + o;               o += (size_t)N_EDGES * HEADS;   // 6.4M
    float*    out1   = w + o;               o += (size_t)N_NODES * IN_CH;   // 6.4M
    float*    h2     = w + o;               o += (size_t)N_NODES * OUT_CH;
    float*    als2   = w + o;               o += (size_t)N_NODES;
    float*    ald2   = w + o;               o += (size_t)N_NODES;
    unsigned* m2raw  = (unsigned*)(w + o);
    float*    m2     = w + o;               o += (size_t)N_NODES;
    float*    den2   = w + o;               o += (size_t)N_NODES;
    float*    alpha2 = w + o;               o += (size_t)N_EDGES;
    float*    out2   = w + o;               o += (size_t)N_NODES * OUT_CH;
    float*    cws    = w + o;               o += 16;

    const int TB = 256;
    auto blocks = [](long long n, int tb) { return (int)((n + tb - 1) / tb); };

    // init + constants
    k_init<<<blocks((long long)N_NODES * IN_CH, TB), TB, 0, stream>>>(
        out1, out2, m1raw, den1, m2raw, den2);
    k_edge_consts<<<1, 32, 0, stream>>>(We1, ae1, We2, ae2, cws);

    // ---- layer 1 ----
    {   // h1 = x @ W1   [50000,128]x[128,128]; 4 waves / 128-thread block
        int tiles = (N_NODES / 16) * (IN_CH / 16);
        k_gemm_f32<<<blocks(tiles, 4), 128, 0, stream>>>(x, W1, h1, N_NODES, IN_CH, IN_CH);
    }
    k_node_attn<<<blocks((long long)N_NODES * HEADS, TB), TB, 0, stream>>>(
        h1, as1, ad1, als1, ald1, HEADS, HID);
    k_edge_score<<<blocks((long long)N_EDGES * HEADS, TB), TB, 0, stream>>>(
        ei, ew, als1, ald1, cws, 0, alpha1, m1raw, HEADS);
    k_max_fix<<<blocks((long long)N_NODES * HEADS, TB), TB, 0, stream>>>(
        m1raw, m1, N_NODES * HEADS);
    k_edge_exp<<<blocks((long long)N_EDGES * HEADS, TB), TB, 0, stream>>>(
        ei, alpha1, m1, den1, HEADS);
    k_edge_aggr<<<blocks((long long)N_EDGES * IN_CH, TB), TB, 0, stream>>>(
        ei, h1, alpha1, den1, out1, HEADS, HID);

    // ---- layer 2 ----
    {   // h2 = elu(out1 + b1) @ W2   [50000,128]x[128,16]
        int tiles = (N_NODES / 16) * (OUT_CH / 16);
        k_gemm_elu_f32<<<blocks(tiles, 4), 128, 0, stream>>>(
            out1, b1, W2, h2, N_NODES, IN_CH, OUT_CH);
    }
    k_node_attn<<<blocks((long long)N_NODES, TB), TB, 0, stream>>>(
        h2, as2, ad2, als2, ald2, 1, OUT_CH);
    k_edge_score<<<blocks((long long)N_EDGES, TB), TB, 0, stream>>>(
        ei, ew, als2, ald2, cws, HEADS, alpha2, m2raw, 1);
    k_max_fix<<<blocks((long long)N_NODES, TB), TB, 0, stream>>>(m2raw, m2, N_NODES);
    k_edge_exp<<<blocks((long long)N_EDGES, TB), TB, 0, stream>>>(
        ei, alpha2, m2, den2, 1);
    k_edge_aggr<<<blocks((long long)N_EDGES * OUT_CH, TB), TB, 0, stream>>>(
        ei, h2, alpha2, den2, out2, 1, OUT_CH);

    // ---- output ----
    k_logsoftmax<<<blocks((long long)N_NODES, TB), TB, 0, stream>>>(out2, b2, y);
}